// GCNLayer_27685359190673
// MI455X (gfx1250) — compile-verified
//
#include <hip/hip_runtime.h>
#include <hip/hip_bf16.h>

typedef float v2f __attribute__((ext_vector_type(2)));
typedef float v8f __attribute__((ext_vector_type(8)));

#define GB 8
#define GN 8192
#define GE 131072
#define GD 128
#define NROW (GB * GN)          // 65536
#define LDS_STRIDE 132          // 128 + 4 pad: float4-aligned, conflict-free frag reads

// ---------------------------------------------------------------------------
// Kernel A: h = X @ W^T + b  via V_WMMA_F32_16X16X4_F32.
// One block = 16 output rows; 8 waves each own a 16-column slab (8*16 = 128).
// X tile (16x128) staged in LDS and shared by all 8 waves.
// Writes h to BOTH h0 (d_out, gather source) and h1 (d_ws, scatter accumulator
// pre-initialized with h so agg pass needs no zero-fill).
// ---------------------------------------------------------------------------
__global__ __launch_bounds__(256) void gcn_gemm_wmma(
    const float* __restrict__ X, const float* __restrict__ W,
    const float* __restrict__ bias, float* __restrict__ h0,
    float* __restrict__ h1) {
  __shared__ float Xs[16 * LDS_STRIDE];

  const int tid = threadIdx.x;
  const int wave = tid >> 5;          // 0..7  -> column slab
  const int lane = tid & 31;
  const int tileM = blockIdx.x * 16;  // 16 output rows per block

  // Cooperative LDS stage: 16x128 floats, each thread moves 8 floats.
  {
    const int row = tid >> 4;          // 0..15
    const int col = (tid & 15) * 8;    // 0,8,...,120
    const float4* src =
        reinterpret_cast<const float4*>(X + (size_t)(tileM + row) * GD + col);
    float4 a0 = src[0];
    float4 a1 = src[1];
    float* dst = &Xs[row * LDS_STRIDE + col];
    *reinterpret_cast<float4*>(dst) = a0;
    *reinterpret_cast<float4*>(dst + 4) = a1;
  }
  __syncthreads();

  const int colBase = wave * 16;
  const int mrow = lane & 15;          // A: M row / B: N col / D: N col
  const int ksel = (lane >> 4) * 2;    // lanes 16-31 hold K+2,K+3

  // Bias folded into accumulator: every D element in this lane sits at
  // column colBase + (lane&15).
  const float bv = bias[colBase + mrow];
  v8f acc = {bv, bv, bv, bv, bv, bv, bv, bv};

#pragma unroll
  for (int k = 0; k < GD; k += 4) {
    // A fragment: {X[m][k0], X[m][k0+1]}, k0 = k + ksel (ISA 16x4 f32 layout)
    v2f a = *reinterpret_cast<const v2f*>(&Xs[mrow * LDS_STRIDE + k + ksel]);
    // B fragment: B[k'][n] = W[n][k'] -> contiguous float2 in row-major W
    v2f bf = *reinterpret_cast<const v2f*>(
        &W[(size_t)(colBase + mrow) * GD + k + ksel]);
    acc = __builtin_amdgcn_wmma_f32_16x16x4_f32(
        /*neg_a=*/false, a, /*neg_b=*/false, bf,
        /*c_mod=*/(short)0, acc, /*reuse_a=*/false, /*reuse_b=*/false);
  }

  // D layout: VGPR v -> M = v + 8*(lane>>4), N = lane&15
  const int rsel = (lane >> 4) * 8;
#pragma unroll
  for (int v = 0; v < 8; ++v) {
    const size_t idx =
        (size_t)(tileM + v + rsel) * GD + (size_t)(colBase + mrow);
    h0[idx] = acc[v];
    h1[idx] = acc[v];
  }
}

// ---------------------------------------------------------------------------
// Kernel B: for each edge (src,tgt): agg[b,tgt,:] += h[b,src,:] / sqrt(N)
// One wave32 per edge; 4 floats per lane -> 128-wide row.
// unsafeAtomicAdd -> native global_atomic_add_f32 (no CAS expansion).
// ---------------------------------------------------------------------------
__global__ __launch_bounds__(256) void gcn_scatter(
    const float* __restrict__ h, const long long* __restrict__ ei,
    float* __restrict__ agg, float scale) {
  const int wgl = blockIdx.x * 8 + (threadIdx.x >> 5);  // global wave = edge id
  const int lane = threadIdx.x & 31;
  const int b = wgl >> 17;          // E = 2^17
  const int e = wgl & (GE - 1);

  const long long* eb = ei + (size_t)b * 2 * GE;
  const int src = (int)eb[e];
  const int tgt = (int)eb[GE + e];

  const float4 v = *reinterpret_cast<const float4*>(
      h + ((size_t)b * GN + src) * GD + lane * 4);
  float* dst = agg + ((size_t)b * GN + tgt) * GD + lane * 4;

  unsafeAtomicAdd(dst + 0, v.x * scale);
  unsafeAtomicAdd(dst + 1, v.y * scale);
  unsafeAtomicAdd(dst + 2, v.z * scale);
  unsafeAtomicAdd(dst + 3, v.w * scale);
}

// ---------------------------------------------------------------------------
// Kernel C: LayerNorm over D=128 + ReLU + node mask. One wave per row.
// ---------------------------------------------------------------------------
__global__ __launch_bounds__(256) void gcn_ln_relu(
    const float* __restrict__ hin, const float* __restrict__ gamma,
    const float* __restrict__ beta, const float* __restrict__ mask,
    float* __restrict__ out) {
  const int row = blockIdx.x * 8 + (threadIdx.x >> 5);
  const int lane = threadIdx.x & 31;

  const float4 x = *reinterpret_cast<const float4*>(
      hin + (size_t)row * GD + lane * 4);

  float s = x.x + x.y + x.z + x.w;
  float sq = x.x * x.x + x.y * x.y + x.z * x.z + x.w * x.w;
#pragma unroll
  for (int m = 16; m > 0; m >>= 1) {
    s += __shfl_xor(s, m, 32);
    sq += __shfl_xor(sq, m, 32);
  }
  const float mu = s * (1.0f / GD);
  const float var = sq * (1.0f / GD) - mu * mu;
  const float inv = rsqrtf(var + 1e-5f);
  const float mk = mask[row];

  const float4 g = *reinterpret_cast<const float4*>(gamma + lane * 4);
  const float4 be = *reinterpret_cast<const float4*>(beta + lane * 4);

  float4 y;
  y.x = fmaxf((x.x - mu) * inv * g.x + be.x, 0.0f) * mk;
  y.y = fmaxf((x.y - mu) * inv * g.y + be.y, 0.0f) * mk;
  y.z = fmaxf((x.z - mu) * inv * g.z + be.z, 0.0f) * mk;
  y.w = fmaxf((x.w - mu) * inv * g.w + be.w, 0.0f) * mk;

  *reinterpret_cast<float4*>(out + (size_t)row * GD + lane * 4) = y;
}

extern "C" void kernel_launch(void* const* d_in, const int* in_sizes, int n_in,
                              void* d_out, int out_size, void* d_ws,
                              size_t ws_size, hipStream_t stream) {
  const float* node_features = (const float*)d_in[0];      // [B,N,128]
  const long long* edge_index = (const long long*)d_in[1]; // [B,2,E] int64
  const float* node_mask = (const float*)d_in[2];          // [B,N]
  const float* W = (const float*)d_in[3];                  // [128,128]
  const float* bias = (const float*)d_in[4];               // [128]
  const float* gamma = (const float*)d_in[5];              // [128]
  const float* beta = (const float*)d_in[6];               // [128]

  float* h = (float*)d_out;      // GEMM result, gather source, final output
  float* acc = (float*)d_ws;     // h + agg/sqrt(N) accumulator (init = h)

  const float scale = 1.0f / sqrtf((float)GN);  // 1/sqrt(8192)

  // A: 65536 rows / 16 per block
  gcn_gemm_wmma<<<NROW / 16, 256, 0, stream>>>(node_features, W, bias, h, acc);
  // B: B*E = 1,048,576 edges, 8 waves (edges) per block
  gcn_scatter<<<(GB * GE) / 8, 256, 0, stream>>>(h, edge_index, acc, scale);
  // C: 65536 rows, 8 rows per block
  gcn_ln_relu<<<NROW / 8, 256, 0, stream>>>(acc, gamma, beta, node_mask, h);
}